// ObjBaseTransformer_85289460564031
// MI455X (gfx1250) — compile-verified
//
#include <hip/hip_runtime.h>
#include <hip/hip_bf16.h>
#include <math.h>

typedef __attribute__((ext_vector_type(2))) float v2f;
typedef __attribute__((ext_vector_type(8))) float v8f;

#define WMMA4(a,b,c) __builtin_amdgcn_wmma_f32_16x16x4_f32(false,(a),false,(b),(short)0,(c),false,false)

// ---------------------------------------------------------------------------
// Generic fp32 WMMA GEMM: out = epilogue(A[M,K] @ W[K,N] + bias [+ resid])
// LDS staging uses CDNA5 async global->LDS DMA (ASYNCcnt-tracked).
// amode: 0 = row-major A (stride lda)
//        1 = gathered rows: A row r = A[rowidx[2*r]]  (pair_idx gather)
//        2 = union_feat strided: A(r,k) = A[(r/49)*12544 + k*49 + (r%49)]
// emode: bit0 = relu, bit1 = scatter addressing for out/resid:
//        addr(r,c) = (r/49)*12544 + c*49 + (r%49)
// Requires: M % 64 == 0, K % 16 == 0. N arbitrary (guarded).
// ---------------------------------------------------------------------------
__global__ __launch_bounds__(256)
void gemm_f32_wmma(const float* __restrict__ A, const float* __restrict__ W,
                   const float* __restrict__ bias, const float* __restrict__ resid,
                   float* __restrict__ out,
                   int M, int N, int K, int lda, int ldout,
                   const int* __restrict__ rowidx, int amode, int emode)
{
  // As row stride 20 floats (80B): async 16B groups stay aligned+contiguous,
  // and 20*r mod 64 is distinct for r in [0,16) -> conflict-free b64 reads.
  __shared__ float As[64][20];
  __shared__ float Bs[16][64];

  const int t = threadIdx.x;
  const int blockRow = blockIdx.y * 64;
  const int blockCol = blockIdx.x * 64;

  // staging coordinates
  const int ar  = t >> 2;          // 0..63  (A stage row)
  const int ac  = (t & 3) * 4;     // 0,4,8,12
  const int bkr = t >> 4;          // 0..15  (B stage k-row)
  const int bc  = (t & 15) * 4;    // 0..60

  const long arow = blockRow + ar;
  const float* aptr = nullptr;
  long ufbase = 0;
  if (amode == 0) {
    aptr = A + (size_t)arow * lda;
  } else if (amode == 1) {
    aptr = A + (size_t)rowidx[2 * arow] * lda;
  } else { // amode == 2
    long n = arow / 49; long hw = arow % 49;
    ufbase = n * 12544 + hw;
  }

  const int w    = t >> 5;
  const int lane = t & 31;
  const int hi   = lane >> 4;
  const int ln   = lane & 15;
  const int r0   = (w & 3) * 16;   // wave row tile within block
  const int c0   = (w >> 2) * 32;  // wave col pair within block

  v8f acc0 = {}; v8f acc1 = {};

  for (int kb = 0; kb < K; kb += 16) {
    __syncthreads();               // previous tile fully consumed
    // ---- async stage A (64x16) : global -> LDS DMA ----
    if (amode == 2) {
      #pragma unroll
      for (int j = 0; j < 4; j++) {
        const unsigned lds = (unsigned)(size_t)&As[ar][ac + j];
        const float* g = A + ufbase + (size_t)(kb + ac + j) * 49;
        asm volatile("global_load_async_to_lds_b32 %0, %1, off"
                     :: "v"(lds), "v"(g) : "memory");
      }
    } else {
      const unsigned lds = (unsigned)(size_t)&As[ar][ac];
      const float* g = aptr + kb + ac;
      asm volatile("global_load_async_to_lds_b128 %0, %1, off"
                   :: "v"(lds), "v"(g) : "memory");
    }
    // ---- stage B (16x64): async when fully in-bounds, guarded fallback ----
    {
      const int gc = blockCol + bc;
      const float* wrow = W + (size_t)(kb + bkr) * N;
      if (gc + 3 < N) {
        const unsigned lds = (unsigned)(size_t)&Bs[bkr][bc];
        asm volatile("global_load_async_to_lds_b128 %0, %1, off"
                     :: "v"(lds), "v"(wrow + gc) : "memory");
      } else {
        #pragma unroll
        for (int j = 0; j < 4; j++)
          Bs[bkr][bc + j] = (gc + j < N) ? wrow[gc + j] : 0.f;
      }
      if (kb + 16 < K && gc < N)
        __builtin_prefetch(W + (size_t)(kb + 16 + bkr) * N + gc, 0, 0);
    }
    asm volatile("s_wait_asynccnt 0" ::: "memory");
    __syncthreads();               // all waves' DMA landed
    // ---- 16x16x4 WMMA over this K-slab ----
    #pragma unroll
    for (int kk = 0; kk < 4; kk++) {
      const int kc = kk * 4 + 2 * hi;
      v2f a  = { As[r0 + ln][kc],        As[r0 + ln][kc + 1] };
      v2f b0 = { Bs[kc][c0 + ln],        Bs[kc + 1][c0 + ln] };
      v2f b1 = { Bs[kc][c0 + 16 + ln],   Bs[kc + 1][c0 + 16 + ln] };
      acc0 = WMMA4(a, b0, acc0);
      acc1 = WMMA4(a, b1, acc1);
    }
  }

  // ---- epilogue: C layout -> element (m = g + 8*hi, n = ln) ----
  #pragma unroll
  for (int tile = 0; tile < 2; tile++) {
    v8f accv = tile ? acc1 : acc0;
    const int col = blockCol + c0 + tile * 16 + ln;
    if (col >= N) continue;
    const float bi = bias ? bias[col] : 0.f;
    #pragma unroll
    for (int g = 0; g < 8; g++) {
      const int row = blockRow + r0 + g + 8 * hi;
      float v = accv[g] + bi;
      size_t oaddr;
      if (emode & 2) {
        const int n = row / 49, hw = row % 49;
        oaddr = (size_t)n * 12544 + (size_t)col * 49 + hw;
      } else {
        oaddr = (size_t)row * ldout + col;
      }
      if (resid) v += resid[oaddr];
      if (emode & 1) v = fmaxf(v, 0.f);
      out[oaddr] = v;
    }
  }
}

// ---------------------------------------------------------------------------
// 3x3 SAME conv (2 -> 256 channels) on 7x7 masks; writes conv+conv_b into the
// fused buffer in [n][o*49+hw] layout (the vr-GEMM input layout).
// ---------------------------------------------------------------------------
__global__ __launch_bounds__(256)
void conv_kernel(const float* __restrict__ sm, const float* __restrict__ cw,
                 const float* __restrict__ cb, float* __restrict__ fused)
{
  const long idx = (long)blockIdx.x * 256 + threadIdx.x; // 2048*256*49
  if (idx >= (long)2048 * 256 * 49) return;
  const int hw = (int)(idx % 49);
  long tmp = idx / 49;
  const int o = (int)(tmp % 256);
  const int n = (int)(tmp / 256);
  const int y = hw / 7, x = hw % 7;
  float s = cb[o];
  #pragma unroll
  for (int ci = 0; ci < 2; ci++)
    #pragma unroll
    for (int ky = 0; ky < 3; ky++)
      #pragma unroll
      for (int kx = 0; kx < 3; kx++) {
        const int iy = y + ky - 1, ix = x + kx - 1;
        if (iy >= 0 && iy < 7 && ix >= 0 && ix < 7)
          s += sm[(size_t)n * 98 + ci * 49 + iy * 7 + ix] *
               cw[(size_t)o * 18 + ci * 9 + ky * 3 + kx];
      }
  fused[(size_t)n * 12544 + (size_t)o * 49 + hw] = s;
}

// ---------------------------------------------------------------------------
// Embedding gather into x[:,1536:1936] + mask metadata (frame id, obj class).
// ---------------------------------------------------------------------------
__global__ __launch_bounds__(256)
void emb_meta_kernel(const int* __restrict__ pair, const float* __restrict__ boxes,
                     const int* __restrict__ labels,
                     const float* __restrict__ emb1, const float* __restrict__ emb2,
                     float* __restrict__ x, int* __restrict__ frame,
                     int* __restrict__ cls)
{
  const int n = blockIdx.x;
  const int t = threadIdx.x;
  const int si = pair[2 * n], oi = pair[2 * n + 1];
  if (t == 0) {
    frame[n] = (int)boxes[(size_t)oi * 5];
    cls[n]   = labels[oi];
  }
  if (t < 200) {
    x[(size_t)n * 1936 + 1536 + t] = emb1[(size_t)labels[si] * 200 + t];
    x[(size_t)n * 1936 + 1736 + t] = emb2[(size_t)labels[oi] * 200 + t];
  }
}

// ---------------------------------------------------------------------------
// Split qkv (2048 x 5808) into per-head padded q/k/v: [8][2048][256], zero pad.
// ---------------------------------------------------------------------------
__global__ __launch_bounds__(256)
void repack_qkv(const float* __restrict__ qkv, float* __restrict__ qp,
                float* __restrict__ kp, float* __restrict__ vp)
{
  long idx = (long)blockIdx.x * 256 + threadIdx.x;   // 3*8*2048*256
  const int d = (int)(idx & 255); idx >>= 8;
  const int n = (int)(idx & 2047); idx >>= 11;
  const int h = (int)(idx & 7);
  const int which = (int)(idx >> 3);
  float v = 0.f;
  if (d < 242)
    v = qkv[(size_t)n * 5808 + (size_t)which * 1936 + h * 242 + d];
  float* dst = (which == 0) ? qp : (which == 1) ? kp : vp;
  dst[((size_t)h * 2048 + n) * 256 + d] = v;
}

// ---------------------------------------------------------------------------
// Merge per-head attention output [8][2048][256] -> [2048][1936].
// ---------------------------------------------------------------------------
__global__ __launch_bounds__(256)
void merge_heads(const float* __restrict__ op, float* __restrict__ ocat)
{
  const long idx = (long)blockIdx.x * 256 + threadIdx.x; // 2048*1936
  if (idx >= (long)2048 * 1936) return;
  const int c = (int)(idx % 1936);
  const int n = (int)(idx / 1936);
  const int h = c / 242, d = c % 242;
  ocat[idx] = op[((size_t)h * 2048 + n) * 256 + d];
}

// ---------------------------------------------------------------------------
// Flash attention, fp32 WMMA 16x16x4. One workgroup = 64 query rows of one
// head (4 waves x 16 rows). Online softmax; mask bias applied in C layout.
// mode 0: sp (frame equality).  mode 1: tm (class equality AND causal).
// Per-wave causal trip counts diverge -> no workgroup barriers, direct loads.
// ---------------------------------------------------------------------------
__global__ __launch_bounds__(128)
void flash_attn(const float* __restrict__ qp, const float* __restrict__ kp,
                const float* __restrict__ vp, float* __restrict__ op,
                const int* __restrict__ meta, int mode)
{
  __shared__ float pbuf[4][16][18];

  const int t = threadIdx.x;
  const int w = t >> 5;
  const int lane = t & 31;
  const int hi = lane >> 4;
  const int ln = lane & 15;
  const int h = blockIdx.y;
  const int qb = blockIdx.x * 64 + w * 16;

  const float* Q  = qp + ((size_t)h * 2048 + qb) * 256;
  const float* Kb = kp + (size_t)h * 2048 * 256;
  const float* Vb = vp + (size_t)h * 2048 * 256;

  v8f o_acc[16];
  #pragma unroll
  for (int i = 0; i < 16; i++) { v8f z = {}; o_acc[i] = z; }

  float mrow[8], lrow[8];
  int qmeta[8];
  #pragma unroll
  for (int g = 0; g < 8; g++) {
    mrow[g] = -1e30f; lrow[g] = 0.f;
    qmeta[g] = meta[qb + g + 8 * hi];
  }

  const float scale = 1.0f / sqrtf(242.0f);
  const float* qrowptr = Q + (size_t)ln * 256 + 2 * hi;
  const int ktiles = (mode == 1) ? (qb >> 4) + 1 : 128;   // causal skip for tm

  for (int kt = 0; kt < ktiles; kt++) {
    const int kb = kt * 16;

    // ---- S = Q @ K^T over padded 256 dims ----
    v8f s = {};
    const float* krowptr = Kb + (size_t)(kb + ln) * 256 + 2 * hi;
    #pragma unroll 8
    for (int kc = 0; kc < 64; kc++) {
      v2f a = *(const v2f*)(qrowptr + kc * 4);
      v2f b = *(const v2f*)(krowptr + kc * 4);
      s = WMMA4(a, b, s);
    }

    // ---- mask bias + online softmax (row = g + 8*hi, key = kb + ln) ----
    const int km = meta[kb + ln];
    const int kidx = kb + ln;
    float corr[8];
    #pragma unroll
    for (int g = 0; g < 8; g++) {
      const int row = qb + g + 8 * hi;
      const bool allowed = (mode == 1) ? (km == qmeta[g] && kidx <= row)
                                       : (km == qmeta[g]);
      float val = s[g] * scale + (allowed ? 0.f : -1e9f);
      float rm = val;
      rm = fmaxf(rm, __shfl_xor(rm, 1));
      rm = fmaxf(rm, __shfl_xor(rm, 2));
      rm = fmaxf(rm, __shfl_xor(rm, 4));
      rm = fmaxf(rm, __shfl_xor(rm, 8));
      const float nm = fmaxf(mrow[g], rm);
      float p = __expf(val - nm);
      float rs = p;
      rs += __shfl_xor(rs, 1); rs += __shfl_xor(rs, 2);
      rs += __shfl_xor(rs, 4); rs += __shfl_xor(rs, 8);
      const float c = __expf(mrow[g] - nm);
      lrow[g] = lrow[g] * c + rs;
      mrow[g] = nm;
      corr[g] = c;
      s[g] = p;
    }

    // ---- rescale accumulators ----
    #pragma unroll
    for (int tt = 0; tt < 16; tt++)
      #pragma unroll
      for (int g = 0; g < 8; g++)
        o_acc[tt][g] *= corr[g];

    // ---- P: C layout -> A layout via per-wave LDS tile ----
    #pragma unroll
    for (int g = 0; g < 8; g++)
      pbuf[w][g + 8 * hi][ln] = s[g];
    __builtin_amdgcn_wave_barrier();
    v2f pa[4];
    #pragma unroll
    for (int kk = 0; kk < 4; kk++) {
      pa[kk].x = pbuf[w][ln][kk * 4 + 2 * hi];
      pa[kk].y = pbuf[w][ln][kk * 4 + 2 * hi + 1];
    }
    __builtin_amdgcn_wave_barrier();

    // ---- O += P @ V  (16 DH tiles) ----
    const float* vrow0 = Vb + (size_t)(kb + 2 * hi) * 256 + ln;
    #pragma unroll
    for (int tt = 0; tt < 16; tt++) {
      const float* vc = vrow0 + tt * 16;
      #pragma unroll
      for (int kk = 0; kk < 4; kk++) {
        v2f b;
        b.x = vc[(size_t)(kk * 4) * 256];
        b.y = vc[(size_t)(kk * 4 + 1) * 256];
        o_acc[tt] = WMMA4(pa[kk], b, o_acc[tt]);
      }
    }
  }

  // ---- normalize and store (padded cols 242..255 are harmless zeros) ----
  float* Optr = op + ((size_t)h * 2048 + qb) * 256;
  #pragma unroll
  for (int g = 0; g < 8; g++) {
    const float inv = 1.f / lrow[g];
    const int row = g + 8 * hi;
    #pragma unroll
    for (int tt = 0; tt < 16; tt++)
      Optr[(size_t)row * 256 + tt * 16 + ln] = o_acc[tt][g] * inv;
  }
}

// ---------------------------------------------------------------------------
// LayerNorm over D=1936 per row.
// ---------------------------------------------------------------------------
__global__ __launch_bounds__(256)
void ln_kernel(const float* __restrict__ in, const float* __restrict__ g,
               const float* __restrict__ b, float* __restrict__ out)
{
  __shared__ float red[256];
  const int row = blockIdx.x;
  const int t = threadIdx.x;
  const float* x = in + (size_t)row * 1936;

  float s = 0.f;
  for (int c = t; c < 1936; c += 256) s += x[c];
  red[t] = s; __syncthreads();
  for (int o = 128; o > 0; o >>= 1) { if (t < o) red[t] += red[t + o]; __syncthreads(); }
  const float mean = red[0] / 1936.f;
  __syncthreads();

  float v = 0.f;
  for (int c = t; c < 1936; c += 256) { const float d = x[c] - mean; v += d * d; }
  red[t] = v; __syncthreads();
  for (int o = 128; o > 0; o >>= 1) { if (t < o) red[t] += red[t + o]; __syncthreads(); }
  const float rstd = rsqrtf(red[0] / 1936.f + 1e-5f);

  for (int c = t; c < 1936; c += 256)
    out[(size_t)row * 1936 + c] = g[c] * (x[c] - mean) * rstd + b[c];
}

// ---------------------------------------------------------------------------
extern "C" void kernel_launch(void* const* d_in, const int* in_sizes, int n_in,
                              void* d_out, int out_size, void* d_ws, size_t ws_size,
                              hipStream_t stream)
{
  (void)in_sizes; (void)n_in; (void)out_size; (void)ws_size;

  const float* features   = (const float*)d_in[0];
  const int*   pair       = (const int*)  d_in[1];
  const float* boxes      = (const float*)d_in[2];
  const int*   labels     = (const int*)  d_in[3];
  const float* union_feat = (const float*)d_in[4];
  const float* smasks     = (const float*)d_in[5];
  const float* subj_w     = (const float*)d_in[6];
  const float* subj_b     = (const float*)d_in[7];
  const float* obj_w      = (const float*)d_in[8];
  const float* obj_b      = (const float*)d_in[9];
  const float* uf_w       = (const float*)d_in[10];
  const float* uf_b       = (const float*)d_in[11];
  const float* conv_w     = (const float*)d_in[12];
  const float* conv_b     = (const float*)d_in[13];
  const float* vr_w       = (const float*)d_in[14];
  const float* vr_b       = (const float*)d_in[15];
  const float* emb1       = (const float*)d_in[16];
  const float* emb2       = (const float*)d_in[17];
  const float* sp_in_w    = (const float*)d_in[18];
  const float* sp_in_b    = (const float*)d_in[19];
  const float* sp_out_w   = (const float*)d_in[20];
  const float* sp_out_b   = (const float*)d_in[21];
  const float* sp_ln1_g   = (const float*)d_in[22];
  const float* sp_ln1_b   = (const float*)d_in[23];
  const float* sp_ff1_w   = (const float*)d_in[24];
  const float* sp_ff1_b   = (const float*)d_in[25];
  const float* sp_ff2_w   = (const float*)d_in[26];
  const float* sp_ff2_b   = (const float*)d_in[27];
  const float* sp_ln2_g   = (const float*)d_in[28];
  const float* sp_ln2_b   = (const float*)d_in[29];
  const float* tm_in_w    = (const float*)d_in[30];
  const float* tm_in_b    = (const float*)d_in[31];
  const float* tm_out_w   = (const float*)d_in[32];
  const float* tm_out_b   = (const float*)d_in[33];
  const float* tm_ln1_g   = (const float*)d_in[34];
  const float* tm_ln1_b   = (const float*)d_in[35];
  const float* tm_ff1_w   = (const float*)d_in[36];
  const float* tm_ff1_b   = (const float*)d_in[37];
  const float* tm_ff2_w   = (const float*)d_in[38];
  const float* tm_ff2_b   = (const float*)d_in[39];
  const float* tm_ln2_g   = (const float*)d_in[40];
  const float* tm_ln2_b   = (const float*)d_in[41];

  const size_t NTOK = 2048, D = 1936, TD = 3 * D, DFF = 2048;

  // ---------------- workspace layout (floats) ----------------
  float* ws = (float*)d_ws;
  // region A: fused (2048*12544) early, later qkv + qp/kp/vp (disjoint in time)
  float* fused = ws;                               // 25,690,112 floats
  float* qkvb  = ws;                               // 2048*5808 = 11,894,784
  float* qp    = ws + (size_t)NTOK * TD;           // 8*2048*256 each
  float* kp    = qp + (size_t)8 * 2048 * 256;
  float* vp    = kp + (size_t)8 * 2048 * 256;
  size_t off   = (size_t)2048 * 12544;             // end of region A
  float* xbuf  = ws + off; off += NTOK * D;
  float* ybuf  = ws + off; off += NTOK * D;
  float* x2b   = ws + off; off += NTOK * D;
  float* ocat  = ws + off; off += NTOK * D;
  float* ffb   = ws + off; off += NTOK * DFF;
  float* opb   = ws + off; off += (size_t)8 * 2048 * 256;
  int* frame   = (int*)(ws + off); off += 2048;
  int* cls     = (int*)(ws + off); off += 2048;

  const dim3 blk256(256), blk128(128);
  #define GEMM_GRID(M, N) dim3(((N) + 63) / 64, (M) / 64)

  // ---------------- front-end ----------------
  emb_meta_kernel<<<dim3(2048), blk256, 0, stream>>>(pair, boxes, labels, emb1,
                                                     emb2, xbuf, frame, cls);
  conv_kernel<<<dim3((2048u * 256 * 49) / 256), blk256, 0, stream>>>(
      smasks, conv_w, conv_b, fused);
  // subj / obj projections (gathered rows of features) -> x[:,0:512], x[:,512:1024]
  gemm_f32_wmma<<<GEMM_GRID(2048, 512), blk256, 0, stream>>>(
      features, subj_w, subj_b, nullptr, xbuf, 2048, 512, 2048, 2048, (int)D,
      pair + 0, 1, 0);
  gemm_f32_wmma<<<GEMM_GRID(2048, 512), blk256, 0, stream>>>(
      features, obj_w, obj_b, nullptr, xbuf + 512, 2048, 512, 2048, 2048, (int)D,
      pair + 1, 1, 0);
  // union-feature channel mix (+bias +conv residual) scattered into fused layout
  gemm_f32_wmma<<<GEMM_GRID(2048 * 49, 256), blk256, 0, stream>>>(
      union_feat, uf_w, uf_b, fused, fused, 2048 * 49, 256, 256, 0, 0,
      nullptr, 2, 2);
  // vr projection -> x[:,1024:1536]
  gemm_f32_wmma<<<GEMM_GRID(2048, 512), blk256, 0, stream>>>(
      fused, vr_w, vr_b, nullptr, xbuf + 1024, 2048, 512, 12544, 12544, (int)D,
      nullptr, 0, 0);

  // ---------------- encoder layers (1 sp + 3 tm) ----------------
  for (int l = 0; l < 4; l++) {
    const int tl = l - 1;
    const float* w_in  = (l == 0) ? sp_in_w  : tm_in_w  + (size_t)tl * D * TD;
    const float* b_in  = (l == 0) ? sp_in_b  : tm_in_b  + (size_t)tl * TD;
    const float* w_out = (l == 0) ? sp_out_w : tm_out_w + (size_t)tl * D * D;
    const float* b_out = (l == 0) ? sp_out_b : tm_out_b + (size_t)tl * D;
    const float* g1    = (l == 0) ? sp_ln1_g : tm_ln1_g + (size_t)tl * D;
    const float* b1    = (l == 0) ? sp_ln1_b : tm_ln1_b + (size_t)tl * D;
    const float* wf1   = (l == 0) ? sp_ff1_w : tm_ff1_w + (size_t)tl * D * DFF;
    const float* bf1   = (l == 0) ? sp_ff1_b : tm_ff1_b + (size_t)tl * DFF;
    const float* wf2   = (l == 0) ? sp_ff2_w : tm_ff2_w + (size_t)tl * DFF * D;
    const float* bf2   = (l == 0) ? sp_ff2_b : tm_ff2_b + (size_t)tl * D;
    const float* g2    = (l == 0) ? sp_ln2_g : tm_ln2_g + (size_t)tl * D;
    const float* b2    = (l == 0) ? sp_ln2_b : tm_ln2_b + (size_t)tl * D;

    // qkv projection
    gemm_f32_wmma<<<GEMM_GRID(2048, (int)TD), blk256, 0, stream>>>(
        xbuf, w_in, b_in, nullptr, qkvb, 2048, (int)TD, (int)D, (int)D, (int)TD,
        nullptr, 0, 0);
    // split into per-head padded q/k/v
    repack_qkv<<<dim3((3u * 8 * 2048 * 256) / 256), blk256, 0, stream>>>(
        qkvb, qp, kp, vp);
    // masked flash attention
    flash_attn<<<dim3(32, 8), blk128, 0, stream>>>(
        qp, kp, vp, opb, (l == 0) ? frame : cls, (l == 0) ? 0 : 1);
    // merge heads
    merge_heads<<<dim3((unsigned)((2048u * 1936 + 255) / 256)), blk256, 0, stream>>>(
        opb, ocat);
    // output projection + residual
    gemm_f32_wmma<<<GEMM_GRID(2048, (int)D), blk256, 0, stream>>>(
        ocat, w_out, b_out, xbuf, ybuf, 2048, (int)D, (int)D, (int)D, (int)D,
        nullptr, 0, 0);
    ln_kernel<<<dim3(2048), blk256, 0, stream>>>(ybuf, g1, b1, x2b);
    // ffn
    gemm_f32_wmma<<<GEMM_GRID(2048, (int)DFF), blk256, 0, stream>>>(
        x2b, wf1, bf1, nullptr, ffb, 2048, (int)DFF, (int)D, (int)D, (int)DFF,
        nullptr, 0, 1 /*relu*/);
    gemm_f32_wmma<<<GEMM_GRID(2048, (int)D), blk256, 0, stream>>>(
        ffb, wf2, bf2, x2b, ybuf, 2048, (int)D, (int)DFF, (int)DFF, (int)D,
        nullptr, 0, 0);
    ln_kernel<<<dim3(2048), blk256, 0, stream>>>(
        ybuf, g2, b2, (l == 3) ? (float*)d_out : xbuf);
  }
  #undef GEMM_GRID
}